// GravNetBlock_12575664243247
// MI455X (gfx1250) — compile-verified
//
#include <hip/hip_runtime.h>
#include <math.h>

typedef __attribute__((ext_vector_type(2))) float v2f;
typedef __attribute__((ext_vector_type(8))) float v8f;

constexpr int Bn = 32, Pn = 1000, Kn = 40;
constexpr int Nn = Bn * Pn;          // 32000 nodes
constexpr float EPSv = 1e-5f;

__device__ __forceinline__ float elu1(float v) { return v > 0.f ? v : __expf(v) - 1.f; }

// ---------------------------------------------------------------------------
// Pack [out1_w; out2_w] into W96 (96x32) and post_w1 zero-padded into W68 (68x32)
// ---------------------------------------------------------------------------
__global__ void pack_weights_k(const float* __restrict__ out1_w,
                               const float* __restrict__ out2_w,
                               const float* __restrict__ post_w1,
                               float* __restrict__ W96, float* __restrict__ W68)
{
    for (int i = threadIdx.x; i < 96 * 32; i += blockDim.x) {
        int r = i >> 5, c = i & 31;
        W96[i] = (r < 32) ? out1_w[r * 32 + c] : out2_w[(r - 32) * 32 + c];
    }
    for (int i = threadIdx.x; i < 68 * 32; i += blockDim.x) {
        int r = i >> 5, c = i & 31;
        W68[i] = (r < 67) ? post_w1[r * 32 + c] : 0.f;
    }
}

// ---------------------------------------------------------------------------
// Generic WMMA GEMM: Y[nrows x 32] = act(X[nrows x kdim] @ W[kdim x 32] + bias)
// One wave handles a 16-row strip (two 16x16 col tiles). kdim % 4 == 0,
// nrows % 16 == 0 (all call sites satisfy this -> EXEC always all-ones).
// Uses V_WMMA_F32_16X16X4_F32 per ISA 7.12.2 fragment layouts.
// ---------------------------------------------------------------------------
__global__ void gemm_wmma_n32(const float* __restrict__ X, int ldx, int kdim,
                              const float* __restrict__ W,
                              const float* __restrict__ bias,
                              float* __restrict__ Y, int ldy, int nrows, int act)
{
    int wave = blockIdx.x * (blockDim.x >> 5) + (threadIdx.x >> 5);
    int row0 = wave * 16;
    if (row0 >= nrows) return;

    int t   = threadIdx.x & 31;
    int t16 = t & 15;
    int hi  = t >> 4;                    // 0: K+0/K+1, 1: K+2/K+3 halves

    v8f c0 = {};                         // cols 0..15
    v8f c1 = {};                         // cols 16..31

    const int arow = row0 + t16;
    for (int k0 = 0; k0 < kdim; k0 += 4) {
        int ka = k0 + 2 * hi;
        v2f a;  a.x  = X[(size_t)arow * ldx + ka];
                a.y  = X[(size_t)arow * ldx + ka + 1];
        v2f b0; b0.x = W[(size_t)ka * 32 + t16];
                b0.y = W[(size_t)(ka + 1) * 32 + t16];
        v2f b1; b1.x = W[(size_t)ka * 32 + 16 + t16];
                b1.y = W[(size_t)(ka + 1) * 32 + 16 + t16];
        c0 = __builtin_amdgcn_wmma_f32_16x16x4_f32(false, a, false, b0, (short)0, c0, false, false);
        c1 = __builtin_amdgcn_wmma_f32_16x16x4_f32(false, a, false, b1, (short)0, c1, false, false);
    }

    float bias0 = bias ? bias[t16]      : 0.f;
    float bias1 = bias ? bias[16 + t16] : 0.f;

#pragma unroll
    for (int r = 0; r < 8; ++r) {
        int row = row0 + r + (hi ? 8 : 0);
        float v0 = c0[r] + bias0;
        float v1 = c1[r] + bias1;
        if (act == 1) { v0 = elu1(v0); v1 = elu1(v1); }
        Y[(size_t)row * ldy + t16]      = v0;
        Y[(size_t)row * ldy + 16 + t16] = v1;
    }
}

// ---------------------------------------------------------------------------
// BatchNorm statistics: per-channel sum / sumsq over Nn rows (32 channels)
// ---------------------------------------------------------------------------
__global__ void bn_stats_k(const float* __restrict__ src,
                           float* __restrict__ sum, float* __restrict__ sumsq)
{
    __shared__ float ls[256], ls2[256];
    int tid = threadIdx.x;
    int c = tid & 31, w = tid >> 5;
    float s = 0.f, s2 = 0.f;
    for (int row = blockIdx.x * 8 + w; row < Nn; row += gridDim.x * 8) {
        float v = src[(size_t)row * 32 + c];
        s += v; s2 += v * v;
    }
    ls[tid] = s; ls2[tid] = s2;
    __syncthreads();
    if (w == 0) {
        for (int ww = 1; ww < 8; ++ww) { s += ls[c + 32 * ww]; s2 += ls2[c + 32 * ww]; }
        atomicAdd(&sum[c], s);
        atomicAdd(&sumsq[c], s2);
    }
}

__global__ void bn_finalize_k(const float* __restrict__ sum, const float* __restrict__ sumsq,
                              float* __restrict__ mean, float* __restrict__ rstd)
{
    int c = threadIdx.x;
    if (c < 32) {
        float m = sum[c] * (1.f / (float)Nn);
        float v = sumsq[c] * (1.f / (float)Nn) - m * m;
        mean[c] = m;
        rstd[c] = rsqrtf(v + EPSv);
    }
}

// Apply BN1 and scatter h into the packed concat matrices for later GEMMs.
__global__ void bn_apply_mid_k(const float* __restrict__ src,
                               const float* __restrict__ mean, const float* __restrict__ rstd,
                               const float* __restrict__ g, const float* __restrict__ b,
                               float* __restrict__ h, float* __restrict__ cat96,
                               float* __restrict__ cat68)
{
    int idx = blockIdx.x * blockDim.x + threadIdx.x;
    if (idx >= Nn * 32) return;
    int c = idx & 31, n = idx >> 5;
    float v = (src[idx] - mean[c]) * rstd[c] * g[c] + b[c];
    h[idx] = v;
    cat96[(size_t)n * 96 + c] = v;            // [h | aggr_mean | aggr_max]
    cat68[(size_t)n * 68 + 35 + c] = v;       // [xgn | s | h | pad]
}

__global__ void bn_apply_out_k(const float* __restrict__ src,
                               const float* __restrict__ mean, const float* __restrict__ rstd,
                               const float* __restrict__ g, const float* __restrict__ b,
                               float* __restrict__ out)
{
    int idx = blockIdx.x * blockDim.x + threadIdx.x;
    if (idx >= Nn * 32) return;
    int c = idx & 31;
    out[idx] = (src[idx] - mean[c]) * rstd[c] * g[c] + b[c];
}

// ---------------------------------------------------------------------------
// s = h @ lin_s_w + lin_s_b  (32 -> 3); also fills concat cols 32..34 + pad col
// ---------------------------------------------------------------------------
__global__ void s_kernel(const float* __restrict__ h,
                         const float* __restrict__ lin_s_w, const float* __restrict__ lin_s_b,
                         float* __restrict__ sbuf, float* __restrict__ cat68)
{
    int n = blockIdx.x * blockDim.x + threadIdx.x;
    if (n >= Nn) return;
    const float* hr = h + (size_t)n * 32;
    float s0 = lin_s_b[0], s1 = lin_s_b[1], s2 = lin_s_b[2];
#pragma unroll
    for (int c = 0; c < 32; ++c) {
        float v = hr[c];
        s0 += v * lin_s_w[c * 3 + 0];
        s1 += v * lin_s_w[c * 3 + 1];
        s2 += v * lin_s_w[c * 3 + 2];
    }
    sbuf[(size_t)n * 3 + 0] = s0;
    sbuf[(size_t)n * 3 + 1] = s1;
    sbuf[(size_t)n * 3 + 2] = s2;
    float* cr = cat68 + (size_t)n * 68;
    cr[32] = s0; cr[33] = s1; cr[34] = s2; cr[67] = 0.f;
}

// ---------------------------------------------------------------------------
// Per-event kNN (K=40, includes self) in learned 3D space; coords in LDS.
// grid = Bn * 8 blocks of 128 threads (one query per thread).
// ---------------------------------------------------------------------------
__global__ void knn_kernel(const float* __restrict__ sbuf,
                           int* __restrict__ knn_i, float* __restrict__ knn_w)
{
    __shared__ float sx[Pn], sy[Pn], sz[Pn];
    int ev = blockIdx.x >> 3;
    int chunk = blockIdx.x & 7;
    int tid = threadIdx.x;
    for (int i = tid; i < Pn; i += blockDim.x) {
        const float* p = sbuf + (size_t)(ev * Pn + i) * 3;
        sx[i] = p[0]; sy[i] = p[1]; sz[i] = p[2];
    }
    __syncthreads();
    int q = chunk * 128 + tid;
    if (q >= Pn) return;

    float qx = sx[q], qy = sy[q], qz = sz[q];
    float bd[Kn]; int bi[Kn];
    float wv = -1.f; int wp = 0;
    for (int k = 0; k < Kn; ++k) {
        float dx = sx[k] - qx, dy = sy[k] - qy, dz = sz[k] - qz;
        float d2 = dx * dx + dy * dy + dz * dz;
        bd[k] = d2; bi[k] = k;
        if (d2 > wv) { wv = d2; wp = k; }
    }
    for (int j = Kn; j < Pn; ++j) {
        float dx = sx[j] - qx, dy = sy[j] - qy, dz = sz[j] - qz;
        float d2 = dx * dx + dy * dy + dz * dz;
        if (d2 < wv) {
            bd[wp] = d2; bi[wp] = j;
            wv = -1.f;
            for (int k = 0; k < Kn; ++k)
                if (bd[k] > wv) { wv = bd[k]; wp = k; }
        }
    }
    size_t base = (size_t)(ev * Pn + q) * Kn;
    for (int k = 0; k < Kn; ++k) {
        knn_w[base + k] = __expf(-10.f * bd[k]);
        knn_i[base + k] = bi[k];
    }
}

// ---------------------------------------------------------------------------
// Aggregation: wave per node, lane = channel; mean + max of w*feat over K.
// Writes directly into cat96 cols [32..63]=mean, [64..95]=max.
// ---------------------------------------------------------------------------
__global__ void aggr_kernel(const float* __restrict__ feat,
                            const int* __restrict__ knn_i,
                            const float* __restrict__ knn_w,
                            float* __restrict__ cat96)
{
    int node = blockIdx.x * (blockDim.x >> 5) + (threadIdx.x >> 5);
    if (node >= Nn) return;
    int c = threadIdx.x & 31;
    int evbase = (node / Pn) * Pn;
    float msum = 0.f, mmax = -INFINITY;
    size_t base = (size_t)node * Kn;
    for (int k = 0; k < Kn; ++k) {
        int j = knn_i[base + k];
        float w = knn_w[base + k];
        float v = w * feat[(size_t)(evbase + j) * 32 + c];
        msum += v;
        mmax = fmaxf(mmax, v);
    }
    cat96[(size_t)node * 96 + 32 + c] = msum * (1.f / (float)Kn);
    cat96[(size_t)node * 96 + 64 + c] = mmax;
}

// ---------------------------------------------------------------------------
extern "C" void kernel_launch(void* const* d_in, const int* in_sizes, int n_in,
                              void* d_out, int out_size, void* d_ws, size_t ws_size,
                              hipStream_t stream)
{
    const float* x       = (const float*)d_in[0];
    // d_in[1] original_coords (unused by reference), d_in[2] batch (layout is implicit)
    const float* pre_w1  = (const float*)d_in[3];
    const float* pre_b1  = (const float*)d_in[4];
    const float* pre_w2  = (const float*)d_in[5];
    const float* pre_b2  = (const float*)d_in[6];
    const float* bn1_g   = (const float*)d_in[7];
    const float* bn1_b   = (const float*)d_in[8];
    const float* lin_s_w = (const float*)d_in[9];
    const float* lin_s_b = (const float*)d_in[10];
    const float* lin_h_w = (const float*)d_in[11];
    const float* lin_h_b = (const float*)d_in[12];
    const float* out1_w  = (const float*)d_in[13];
    const float* out2_w  = (const float*)d_in[14];
    const float* out2_b  = (const float*)d_in[15];
    const float* post_w1 = (const float*)d_in[16];
    const float* post_b1 = (const float*)d_in[17];
    const float* post_w2 = (const float*)d_in[18];
    const float* post_b2 = (const float*)d_in[19];
    const float* bn2_g   = (const float*)d_in[20];
    const float* bn2_b   = (const float*)d_in[21];
    float* out = (float*)d_out;

    // Workspace layout (floats)
    float* ws = (float*)d_ws;
    size_t off = 0;
    auto alloc = [&](size_t n) { float* p = ws + off; off += n; return p; };
    float* h2raw  = alloc((size_t)Nn * 32);
    float* h      = alloc((size_t)Nn * 32);
    float* feat   = alloc((size_t)Nn * 32);
    float* p1     = alloc((size_t)Nn * 32);   // pre-MLP hidden, reused as post hidden
    float* outraw = alloc((size_t)Nn * 32);
    float* cat96  = alloc((size_t)Nn * 96);
    float* cat68  = alloc((size_t)Nn * 68);
    float* sbuf   = alloc((size_t)Nn * 3 + 4);
    float* knnw   = alloc((size_t)Nn * Kn);
    int*   knni   = (int*)alloc((size_t)Nn * Kn);
    float* W96    = alloc(96 * 32);
    float* W68    = alloc(68 * 32);
    float* stats  = alloc(128);               // sum[32] sumsq[32] mean[32] rstd[32]
    float* ssum = stats, *ssq = stats + 32, *smean = stats + 64, *srstd = stats + 96;

    const int gemmBlocks = (Nn / 16 + 7) / 8;     // 8 waves / 256-thread block
    const int elemBlocks = (Nn * 32 + 255) / 256;

    pack_weights_k<<<1, 256, 0, stream>>>(out1_w, out2_w, post_w1, W96, W68);

    // pre-MLP: h1 = elu(x @ pre_w1 + b1); h2raw = elu(h1 @ pre_w2 + b2)
    gemm_wmma_n32<<<gemmBlocks, 256, 0, stream>>>(x,  64, 64, pre_w1, pre_b1, p1,    32, Nn, 1);
    gemm_wmma_n32<<<gemmBlocks, 256, 0, stream>>>(p1, 32, 32, pre_w2, pre_b2, h2raw, 32, Nn, 1);

    // BN1
    hipMemsetAsync(ssum, 0, 64 * sizeof(float), stream);
    bn_stats_k<<<64, 256, 0, stream>>>(h2raw, ssum, ssq);
    bn_finalize_k<<<1, 32, 0, stream>>>(ssum, ssq, smean, srstd);
    bn_apply_mid_k<<<elemBlocks, 256, 0, stream>>>(h2raw, smean, srstd, bn1_g, bn1_b, h, cat96, cat68);

    // learned coords + feature projection
    s_kernel<<<(Nn + 255) / 256, 256, 0, stream>>>(h, lin_s_w, lin_s_b, sbuf, cat68);
    gemm_wmma_n32<<<gemmBlocks, 256, 0, stream>>>(h, 32, 32, lin_h_w, lin_h_b, feat, 32, Nn, 0);

    // kNN + weighted mean/max aggregation
    knn_kernel<<<Bn * 8, 128, 0, stream>>>(sbuf, knni, knnw);
    aggr_kernel<<<(Nn + 3) / 4, 128, 0, stream>>>(feat, knni, knnw, cat96);

    // xgn = [h | aggr] @ [out1_w; out2_w] + out2_b  -> cat68 cols 0..31
    gemm_wmma_n32<<<gemmBlocks, 256, 0, stream>>>(cat96, 96, 96, W96, out2_b, cat68, 68, Nn, 0);

    // post-MLP
    gemm_wmma_n32<<<gemmBlocks, 256, 0, stream>>>(cat68, 68, 68, W68, post_b1, p1,     32, Nn, 1);
    gemm_wmma_n32<<<gemmBlocks, 256, 0, stream>>>(p1,    32, 32, post_w2, post_b2, outraw, 32, Nn, 1);

    // BN2 -> final output
    hipMemsetAsync(ssum, 0, 64 * sizeof(float), stream);
    bn_stats_k<<<64, 256, 0, stream>>>(outraw, ssum, ssq);
    bn_finalize_k<<<1, 32, 0, stream>>>(ssum, ssq, smean, srstd);
    bn_apply_out_k<<<elemBlocks, 256, 0, stream>>>(outraw, smean, srstd, bn2_g, bn2_b, out);
}